// SumNodeLabelAggregation_5153960755615
// MI455X (gfx1250) — compile-verified
//
#include <hip/hip_runtime.h>
#include <hip/hip_bf16.h>

#define FEAT 48            // feature dim
#define LPN  12            // lanes per node: 12 * float4 = 48 floats
#define SCAN_BLOCK 1024
#define NODES_PER_BLOCK 16
#define AGG_THREADS (NODES_PER_BLOCK * LPN)   // 192 = 6 waves (wave32)
#define COLS_CAP 1024      // LDS staging capacity (ints) per block; avg need ~256

typedef __attribute__((ext_vector_type(4))) float v4f;
typedef __attribute__((ext_vector_type(4))) int   v4i;

// ---------------- CSR build path ----------------

__global__ void zero_ints_kernel(int* __restrict__ p, int n) {
    int i = blockIdx.x * blockDim.x + threadIdx.x;
    if (i < n) p[i] = 0;
}

// Scalar degree count (generic fallback).
__global__ void count_degrees_kernel(const int* __restrict__ rows, int E,
                                     int* __restrict__ deg) {
    int e = blockIdx.x * blockDim.x + threadIdx.x;
    if (e < E) {
        __builtin_prefetch(rows + e + 8192, 0, 0);
        atomicAdd(&deg[rows[e]], 1);
    }
}

// Vectorized degree count: 4 edges/thread via b128 (requires E % 4 == 0).
__global__ void count_degrees4_kernel(const int* __restrict__ rows, int E,
                                      int* __restrict__ deg) {
    int e4 = (blockIdx.x * blockDim.x + threadIdx.x) * 4;
    if (e4 < E) {
        __builtin_prefetch(rows + e4 + 16384, 0, 0);
        v4i r = *(const v4i*)(rows + e4);
        atomicAdd(&deg[r.x], 1);
        atomicAdd(&deg[r.y], 1);
        atomicAdd(&deg[r.z], 1);
        atomicAdd(&deg[r.w], 1);
    }
}

// Per-block inclusive scan in LDS -> exclusive offsets + block totals.
__global__ __launch_bounds__(SCAN_BLOCK)
void scan_blocks_kernel(const int* __restrict__ deg, int* __restrict__ offs,
                        int* __restrict__ bsums, int n) {
    __shared__ int s[SCAN_BLOCK];
    const int t = threadIdx.x;
    const int i = blockIdx.x * SCAN_BLOCK + t;
    const int v = (i < n) ? deg[i] : 0;
    s[t] = v;
    __syncthreads();
    for (int off = 1; off < SCAN_BLOCK; off <<= 1) {
        int add = (t >= off) ? s[t - off] : 0;
        __syncthreads();
        s[t] += add;
        __syncthreads();
    }
    if (i < n) offs[i] = s[t] - v;                      // exclusive within block
    if (t == SCAN_BLOCK - 1) bsums[blockIdx.x] = s[t];  // block total
}

// Single-block exclusive scan of block totals (nb <= 1024).
__global__ __launch_bounds__(SCAN_BLOCK)
void scan_sums_kernel(int* __restrict__ bsums, int nb) {
    __shared__ int s[SCAN_BLOCK];
    const int t = threadIdx.x;
    const int v = (t < nb) ? bsums[t] : 0;
    s[t] = v;
    __syncthreads();
    for (int off = 1; off < SCAN_BLOCK; off <<= 1) {
        int add = (t >= off) ? s[t - off] : 0;
        __syncthreads();
        s[t] += add;
        __syncthreads();
    }
    if (t < nb) bsums[t] = s[t] - v;
}

__global__ void add_offsets_kernel(int* __restrict__ offs, int* __restrict__ cur,
                                   const int* __restrict__ bsums, int n) {
    int i = blockIdx.x * blockDim.x + threadIdx.x;
    if (i < n) {
        int o = offs[i] + bsums[i / SCAN_BLOCK];
        offs[i] = o;
        cur[i]  = o;   // scatter cursor starts at row offset
    }
}

// Scalar scatter (generic fallback).
__global__ void scatter_edges_kernel(const int* __restrict__ rows,
                                     const int* __restrict__ cols, int E,
                                     int* __restrict__ cur,
                                     int* __restrict__ csr_cols) {
    int e = blockIdx.x * blockDim.x + threadIdx.x;
    if (e < E) {
        int r = __builtin_nontemporal_load(rows + e);   // last use of edge stream
        int c = __builtin_nontemporal_load(cols + e);
        int p = atomicAdd(&cur[r], 1);
        csr_cols[p] = c;
    }
}

// Vectorized scatter: 4 edges/thread via b128 NT loads (requires E % 4 == 0).
__global__ void scatter_edges4_kernel(const int* __restrict__ rows,
                                      const int* __restrict__ cols, int E,
                                      int* __restrict__ cur,
                                      int* __restrict__ csr_cols) {
    int e4 = (blockIdx.x * blockDim.x + threadIdx.x) * 4;
    if (e4 < E) {
        v4i r = __builtin_nontemporal_load((const v4i*)(rows + e4));
        v4i c = __builtin_nontemporal_load((const v4i*)(cols + e4));
        int p0 = atomicAdd(&cur[r.x], 1);
        int p1 = atomicAdd(&cur[r.y], 1);
        int p2 = atomicAdd(&cur[r.z], 1);
        int p3 = atomicAdd(&cur[r.w], 1);
        csr_cols[p0] = c.x;
        csr_cols[p1] = c.y;
        csr_cols[p2] = c.z;
        csr_cols[p3] = c.w;
    }
}

// Atomic-free aggregation. Block = 16 nodes x 12 lanes; each lane owns a
// float4 slice of the 48-dim feature. The 16 nodes' neighbor lists are
// CONTIGUOUS in csr_cols, so we stage the whole slice into LDS with the
// gfx1250 async-to-LDS DMA path, then run a 4-way-unrolled gather loop
// (4 independent L2 gathers in flight per lane; x fits in the 192MB L2).
// Output rows are streamed with non-temporal b128 stores.
__global__ __launch_bounds__(AGG_THREADS)
void aggregate_kernel(const float* __restrict__ x,
                      const int* __restrict__ offs,
                      const int* __restrict__ deg,
                      const int* __restrict__ csr_cols,
                      float* __restrict__ out, int N) {
    __shared__ int scols[COLS_CAP];
    const int tid = threadIdx.x;
    const int n0  = blockIdx.x * NODES_PER_BLOCK;
    const int nLast = (n0 + NODES_PER_BLOCK <= N) ? (n0 + NODES_PER_BLOCK - 1) : (N - 1);

    const int span_start = offs[n0];
    const int span_end   = offs[nLast] + deg[nLast];
    const int len        = span_end - span_start;   // uniform across block

    const bool use_lds = (len <= COLS_CAP);
    if (use_lds) {
        for (int i = tid; i < len; i += AGG_THREADS) {
            // low 32 bits of a generic shared pointer == LDS byte offset
            unsigned lds_off = (unsigned)(uintptr_t)(&scols[i]);
            unsigned long long gaddr =
                (unsigned long long)(uintptr_t)(csr_cols + span_start + i);
            asm volatile("global_load_async_to_lds_b32 %0, %1, off"
                         :: "v"(lds_off), "v"(gaddr) : "memory");
        }
#if __has_builtin(__builtin_amdgcn_s_wait_asynccnt)
        __builtin_amdgcn_s_wait_asynccnt(0);
#else
        asm volatile("s_wait_asynccnt 0x0" ::: "memory");
#endif
    }
    __syncthreads();   // uniform: use_lds depends only on blockIdx

    const int n = n0 + tid / LPN;
    const int t = tid % LPN;
    if (n >= N) return;

    const int start = offs[n];
    const int cnt   = deg[n];

    v4f acc = (v4f)(0.f);
    if (use_lds) {
        const int base = start - span_start;
        int k = 0;
        for (; k + 4 <= cnt; k += 4) {
            int c0 = scols[base + k + 0];
            int c1 = scols[base + k + 1];
            int c2 = scols[base + k + 2];
            int c3 = scols[base + k + 3];
            v4f v0 = *(const v4f*)(x + (size_t)c0 * FEAT + t * 4);
            v4f v1 = *(const v4f*)(x + (size_t)c1 * FEAT + t * 4);
            v4f v2 = *(const v4f*)(x + (size_t)c2 * FEAT + t * 4);
            v4f v3 = *(const v4f*)(x + (size_t)c3 * FEAT + t * 4);
            acc += (v0 + v1) + (v2 + v3);
        }
        for (; k < cnt; ++k) {
            int c = scols[base + k];
            acc += *(const v4f*)(x + (size_t)c * FEAT + t * 4);
        }
    } else {
        for (int k = 0; k < cnt; ++k) {
            int c = csr_cols[start + k];
            acc += *(const v4f*)(x + (size_t)c * FEAT + t * 4);
        }
    }

    const v4f self = *(const v4f*)(x + (size_t)n * FEAT + t * 4);
    float* orow = out + (size_t)n * (2 * FEAT);
    __builtin_nontemporal_store(self, (v4f*)(orow + t * 4));         // out[:, 0:48]
    __builtin_nontemporal_store(acc,  (v4f*)(orow + FEAT + t * 4));  // out[:, 48:96]
}

// ---------------- Fallback: direct f32 atomics (if ws too small) ----------------

__global__ void init_out_kernel(const float* __restrict__ x,
                                float* __restrict__ out, int N) {
    int gid = blockIdx.x * blockDim.x + threadIdx.x;
    int n = gid / LPN;
    int t = gid % LPN;
    if (n >= N) return;
    const v4f self = *(const v4f*)(x + (size_t)n * FEAT + t * 4);
    float* orow = out + (size_t)n * (2 * FEAT);
    *(v4f*)(orow + t * 4)        = self;
    *(v4f*)(orow + FEAT + t * 4) = (v4f)(0.f);
}

__global__ void edge_atomic_kernel(const float* __restrict__ x,
                                   const int* __restrict__ rows,
                                   const int* __restrict__ cols,
                                   float* __restrict__ out, int E) {
    int gid = blockIdx.x * blockDim.x + threadIdx.x;
    int e = gid / LPN;
    int t = gid % LPN;
    if (e >= E) return;
    int r = rows[e];
    int c = cols[e];
    const v4f v = *(const v4f*)(x + (size_t)c * FEAT + t * 4);
    float* p = out + (size_t)r * (2 * FEAT) + FEAT + t * 4;
    // agent-scope relaxed fp32 adds -> global_atomic_add_f32
    __hip_atomic_fetch_add(p + 0, v.x, __ATOMIC_RELAXED, __HIP_MEMORY_SCOPE_AGENT);
    __hip_atomic_fetch_add(p + 1, v.y, __ATOMIC_RELAXED, __HIP_MEMORY_SCOPE_AGENT);
    __hip_atomic_fetch_add(p + 2, v.z, __ATOMIC_RELAXED, __HIP_MEMORY_SCOPE_AGENT);
    __hip_atomic_fetch_add(p + 3, v.w, __ATOMIC_RELAXED, __HIP_MEMORY_SCOPE_AGENT);
}

// ---------------- launch ----------------

extern "C" void kernel_launch(void* const* d_in, const int* in_sizes, int n_in,
                              void* d_out, int out_size, void* d_ws, size_t ws_size,
                              hipStream_t stream) {
    const float* x   = (const float*)d_in[0];
    const int*   ei  = (const int*)d_in[1];   // edge_index, flat [2, E]
    float*       out = (float*)d_out;

    const int N = in_sizes[0] / FEAT;         // 100000
    const int E = in_sizes[1] / 2;            // 1600000
    const int* rows = ei;
    const int* cols = ei + E;

    // Workspace layout (ints): deg[N] | offs[N] | cur[N] | bsums[1024] | csr_cols[E]
    const size_t need = ((size_t)3 * N + SCAN_BLOCK + (size_t)E) * sizeof(int);

    if (ws_size >= need) {
        int* deg      = (int*)d_ws;
        int* offs     = deg + N;
        int* cur      = offs + N;
        int* bsums    = cur + N;
        int* csr_cols = bsums + SCAN_BLOCK;

        const int nb = (N + SCAN_BLOCK - 1) / SCAN_BLOCK;   // 98 <= 1024
        const bool vec4 = ((E & 3) == 0);                   // b128 alignment of cols

        zero_ints_kernel<<<(N + 255) / 256, 256, 0, stream>>>(deg, N);
        if (vec4) {
            const int q = E / 4;
            count_degrees4_kernel<<<(q + 255) / 256, 256, 0, stream>>>(rows, E, deg);
        } else {
            count_degrees_kernel<<<(E + 255) / 256, 256, 0, stream>>>(rows, E, deg);
        }
        scan_blocks_kernel<<<nb, SCAN_BLOCK, 0, stream>>>(deg, offs, bsums, N);
        scan_sums_kernel<<<1, SCAN_BLOCK, 0, stream>>>(bsums, nb);
        add_offsets_kernel<<<(N + 255) / 256, 256, 0, stream>>>(offs, cur, bsums, N);
        if (vec4) {
            const int q = E / 4;
            scatter_edges4_kernel<<<(q + 255) / 256, 256, 0, stream>>>(rows, cols, E, cur, csr_cols);
        } else {
            scatter_edges_kernel<<<(E + 255) / 256, 256, 0, stream>>>(rows, cols, E, cur, csr_cols);
        }

        const int nblocks = (N + NODES_PER_BLOCK - 1) / NODES_PER_BLOCK;
        aggregate_kernel<<<nblocks, AGG_THREADS, 0, stream>>>(
            x, offs, deg, csr_cols, out, N);
    } else {
        // Fallback: direct scatter with hardware fp32 atomics.
        const long long tn = (long long)N * LPN;
        init_out_kernel<<<(int)((tn + 191) / 192), 192, 0, stream>>>(x, out, N);
        const long long te = (long long)E * LPN;
        edge_atomic_kernel<<<(int)((te + 191) / 192), 192, 0, stream>>>(
            x, rows, cols, out, E);
    }
}